// SAGEMeanConv_26783416058446
// MI455X (gfx1250) — compile-verified
//
#include <hip/hip_runtime.h>

typedef __attribute__((ext_vector_type(2))) float v2f;
typedef __attribute__((ext_vector_type(8))) float v8f;

#define IN_FEATS 256
#define OUT_FEATS 128

// ---------------------------------------------------------------------------
// GEMM: h_self[N,128] = feat[N,256] @ W[256,128]  (fp32, V_WMMA_F32_16X16X4_F32)
// One 16-row output tile per block; 8 waves each own one 16-col tile.
// ---------------------------------------------------------------------------
__global__ __launch_bounds__(256) void sage_gemm_wmma(
    const float* __restrict__ feat, const float* __restrict__ Wm,
    float* __restrict__ out, int n)
{
    // A tile in LDS: 16 rows x 256 cols, padded row stride 260 floats.
    // Bank for lane l reading column k: (260*l + k) % 64 = (4l + k) % 64 ->
    // conflict-free across all 32 lanes (halves offset by 2 are disjoint).
    __shared__ float As[16][260];

    const int tid  = threadIdx.x;
    const int row0 = blockIdx.x * 16;

    // Cooperative, coalesced float4 load of the A tile (guarded on rows).
    {
        const int r  = tid >> 4;          // 0..15
        const int c4 = (tid & 15) * 4;    // 0..60
        const bool ok = (row0 + r) < n;
#pragma unroll
        for (int j = 0; j < 4; ++j) {
            const int col = c4 + j * 64;
            float4 v = make_float4(0.f, 0.f, 0.f, 0.f);
            if (ok) v = *(const float4*)(feat + (size_t)(row0 + r) * IN_FEATS + col);
            *(float4*)(&As[r][col]) = v;
        }
    }
    __syncthreads();

    const int wave = tid >> 5;        // 0..7 -> column tile
    const int lane = tid & 31;
    const int half = lane >> 4;       // 0: K pair 0,1 / rows 0-7 ; 1: K pair 2,3 / rows 8-15
    const int l    = lane & 15;
    const int col0 = wave * 16;

    v8f acc = {};

#pragma unroll 8
    for (int kb = 0; kb < IN_FEATS / 4; ++kb) {
        const int k = kb * 4 + half * 2;
        v2f a, b;
        // A 16x4: VGPR0 = K (lanes 0-15: K=0, lanes 16-31: K=2), VGPR1 = K+1
        a.x = As[l][k];
        a.y = As[l][k + 1];
        // B 4x16: row K striped across 16 lanes (N = col0 + l)
        b.x = Wm[(size_t)k       * OUT_FEATS + col0 + l];
        b.y = Wm[(size_t)(k + 1) * OUT_FEATS + col0 + l];
        acc = __builtin_amdgcn_wmma_f32_16x16x4_f32(
            /*neg_a=*/false, a, /*neg_b=*/false, b,
            /*c_mod=*/(short)0, acc, /*reuse_a=*/false, /*reuse_b=*/false);
    }

    // D 16x16 f32: VGPR r -> row r (lanes 0-15) / row 8+r (lanes 16-31)
    float* o = out + (size_t)(row0 + half * 8) * OUT_FEATS + col0 + l;
    if (row0 + 16 <= n) {
        // Fast path: whole tile in-bounds (always taken when n % 16 == 0).
#pragma unroll
        for (int r = 0; r < 8; ++r) o[(size_t)r * OUT_FEATS] = acc[r];
    } else {
#pragma unroll
        for (int r = 0; r < 8; ++r) {
            if (row0 + half * 8 + r < n) o[(size_t)r * OUT_FEATS] = acc[r];
        }
    }
}

// ---------------------------------------------------------------------------
// Workspace zeroing (atomics accumulate; must re-zero every launch).
// n4 = number of float4s; caller pads the region to a multiple of 4 floats.
// ---------------------------------------------------------------------------
__global__ void zero_ws(float4* __restrict__ p, size_t n4)
{
    size_t i = (size_t)blockIdx.x * blockDim.x + threadIdx.x;
    if (i < n4) p[i] = make_float4(0.f, 0.f, 0.f, 0.f);
}

// ---------------------------------------------------------------------------
// Edge scatter: h_neigh[dst[e],:] += h_self[src[e],:]  and  degs[dst[e]] += 1.
// One thread = one float4 of one edge (32 quads per 128-float row); the
// quad-0 thread of each edge also bumps the degree counter, merging the
// degree pass into this kernel (saves a full re-read of dst).
// h_self/h_neigh are L2-resident (25.6 MB each << 192 MB L2).
// ---------------------------------------------------------------------------
__global__ void scatter_add(const int* __restrict__ src, const int* __restrict__ dst,
                            const float* __restrict__ hself, float* __restrict__ hneigh,
                            float* __restrict__ degs, long long total_quads)
{
    long long g = (long long)blockIdx.x * blockDim.x + threadIdx.x;
    if (g >= total_quads) return;
    const int edge = (int)(g >> 5);          // OUT_FEATS/4 == 32 quads per edge
    const int qid  = (int)g & 31;
    const int q    = qid * 4;
    const int d    = dst[edge];
    const float4 v = *(const float4*)(hself + (size_t)src[edge] * OUT_FEATS + q);
    float* o = hneigh + (size_t)d * OUT_FEATS + q;
    atomicAdd(o + 0, v.x);
    atomicAdd(o + 1, v.y);
    atomicAdd(o + 2, v.z);
    atomicAdd(o + 3, v.w);
    if (qid == 0) atomicAdd(&degs[d], 1.0f);
}

// ---------------------------------------------------------------------------
// out = relu((h_self + h_neigh) / (deg + 1)), in place over d_out (= h_self).
// ---------------------------------------------------------------------------
__global__ void finalize(const float* __restrict__ hneigh, const float* __restrict__ degs,
                         float* __restrict__ out, int total)
{
    int i = blockIdx.x * blockDim.x + threadIdx.x;
    if (i < total) {
        const float d = degs[i >> 7] + 1.0f;   // OUT_FEATS == 128
        const float v = (out[i] + hneigh[i]) / d;
        out[i] = v > 0.f ? v : 0.f;
    }
}

extern "C" void kernel_launch(void* const* d_in, const int* in_sizes, int n_in,
                              void* d_out, int out_size, void* d_ws, size_t ws_size,
                              hipStream_t stream)
{
    const float* feat = (const float*)d_in[0];
    const float* Wm   = (const float*)d_in[1];
    const int*   src  = (const int*)d_in[2];
    const int*   dst  = (const int*)d_in[3];
    float* out = (float*)d_out;

    const int n = in_sizes[0] / IN_FEATS;   // 50000
    const int e = in_sizes[2];              // 1600000

    float* hneigh = (float*)d_ws;                       // [n, 128]
    float* degs   = hneigh + (size_t)n * OUT_FEATS;     // [n]

    // 1) zero workspace (h_neigh + degs), float4-vectorized.
    const size_t zfloats = (size_t)n * OUT_FEATS + (size_t)n;
    const size_t z4      = (zfloats + 3) / 4;           // d_ws is 16B-aligned
    zero_ws<<<(int)((z4 + 255) / 256), 256, 0, stream>>>((float4*)d_ws, z4);

    // 2) h_self = feat @ W  ->  d_out  (WMMA)
    sage_gemm_wmma<<<(n + 15) / 16, 256, 0, stream>>>(feat, Wm, out, n);

    // 3) scatter-sum of source rows into h_neigh + fused degree count
    const long long quads = (long long)e * (OUT_FEATS / 4);
    scatter_add<<<(int)((quads + 255) / 256), 256, 0, stream>>>(src, dst, out, hneigh, degs, quads);

    // 4) normalize + ReLU in place on d_out
    const int total = n * OUT_FEATS;
    finalize<<<(total + 255) / 256, 256, 0, stream>>>(hneigh, degs, out, total);
}